// ConditionalLogisticRegression_3667902071369
// MI455X (gfx1250) — compile-verified
//
#include <hip/hip_runtime.h>
#include <math.h>

typedef __attribute__((ext_vector_type(2))) float v2f;
typedef __attribute__((ext_vector_type(8))) float v8f;

#define DIMK 64
#define NSEG 100000
#define TILES_PER_WAVE 4
#define WAVES_PER_BLOCK 8

// ---- order-preserving float <-> uint mapping (for atomicMax on floats) ----
__device__ __forceinline__ unsigned f2o(float f) {
    unsigned u = __float_as_uint(f);
    return (u & 0x80000000u) ? ~u : (u | 0x80000000u);
}
__device__ __forceinline__ float o2f(unsigned u) {
    u = (u & 0x80000000u) ? (u & 0x7FFFFFFFu) : ~u;
    return __uint_as_float(u);
}

// ---- pass 0: zero segment accumulators --------------------------------------
__global__ void init_seg_kernel(unsigned* __restrict__ smax,
                                float* __restrict__ denom, int nseg) {
    int i = blockIdx.x * blockDim.x + threadIdx.x;
    if (i < nseg) { smax[i] = 0u; denom[i] = 0.0f; }
}

// ---- pass 1: logits = X @ w + b via V_WMMA_F32_16X16X4_F32 ------------------
// One wave handles TILES_PER_WAVE tiles of 16 rows. w fragments are hoisted
// into registers once per wave (w is only 256B; X streaming dominates traffic).
__global__ __launch_bounds__(256) void
logits_wmma_kernel(const float* __restrict__ X, const float* __restrict__ w,
                   const float* __restrict__ b, float* __restrict__ logits,
                   int n) {
    const int lane = threadIdx.x & 31;
    const int wave = threadIdx.x >> 5;
    const long tile0 =
        ((long)blockIdx.x * WAVES_PER_BLOCK + wave) * TILES_PER_WAVE;
    if (tile0 * 16 >= n) return;                      // wave-uniform

    const int r    = lane & 15;                       // A row this lane feeds
    const int koff = (lane >> 4) * 2;                 // K sub-offset per lane half
    const int sub  = lane & 7;
    const float bias = b[0];

    // Hoist all B fragments (w broadcast across columns) into registers.
    v2f wfrag[DIMK / 4];
#pragma unroll
    for (int k = 0; k < DIMK / 4; ++k)
        wfrag[k] = *(const v2f*)(w + 4 * k + koff);

#pragma unroll
    for (int t = 0; t < TILES_PER_WAVE; ++t) {
        const long row_base = (tile0 + t) * 16;
        if (row_base >= n) break;                     // wave-uniform

        if (row_base + 16 <= n) {                     // full tile: WMMA path
            const float* xrow = X + (row_base + r) * DIMK;
            v8f acc = {};
#pragma unroll
            for (int k = 0; k < DIMK / 4; ++k) {
                v2f a = *(const v2f*)(xrow + 4 * k + koff);  // 8B aligned
                acc = __builtin_amdgcn_wmma_f32_16x16x4_f32(
                    false, a, false, wfrag[k], (short)0, acc, false, false);
            }
            // Diagonal extraction: all 16 accumulator columns are identical;
            // lanes 0..7 -> rows 0..7 (acc[lane]), lanes 16..23 -> rows 8..15.
            float v = acc[0];
            v = (sub == 1) ? acc[1] : v;
            v = (sub == 2) ? acc[2] : v;
            v = (sub == 3) ? acc[3] : v;
            v = (sub == 4) ? acc[4] : v;
            v = (sub == 5) ? acc[5] : v;
            v = (sub == 6) ? acc[6] : v;
            v = (sub == 7) ? acc[7] : v;
            if ((lane & 15) < 8) {
                long row = row_base + ((lane < 16) ? sub : 8 + sub);
                logits[row] = v + bias;
            }
        } else {                                      // ragged tail: scalar dot
            long row = row_base + lane;
            if (lane < 16 && row < n) {
                float s = 0.0f;
                const float* xrow = X + row * DIMK;
#pragma unroll
                for (int k = 0; k < DIMK; ++k) s += xrow[k] * w[k];
                logits[row] = s + bias;
            }
        }
    }
}

// ---- pass 2: per-segment max, wave-aggregated atomics -----------------------
// seg_ids are sorted, so a wave of 32 consecutive rows spans few segments.
// Segmented suffix-max via shfl_down doubling; run-head lanes issue atomicMax.
__global__ __launch_bounds__(256) void
seg_max_kernel(const float* __restrict__ logits, const int* __restrict__ seg,
               unsigned* __restrict__ smax, int n) {
    const int i    = blockIdx.x * blockDim.x + threadIdx.x;
    const int lane = threadIdx.x & 31;
    const bool valid = (i < n);
    int   s = valid ? seg[i]    : -1;
    float v = valid ? logits[i] : -INFINITY;
#pragma unroll
    for (int d = 1; d < 32; d <<= 1) {
        float ov = __shfl_down(v, (unsigned)d, 32);
        int   os = __shfl_down(s, (unsigned)d, 32);
        if ((lane + d < 32) && (os == s)) v = fmaxf(v, ov);
    }
    int  ps   = __shfl_up(s, 1u, 32);
    bool head = (lane == 0) || (ps != s);
    if (head && valid) atomicMax(&smax[s], f2o(v));
}

// ---- pass 3: z = exp(logit - m); wave-aggregated denom atomicAdd ------------
__global__ __launch_bounds__(256) void
expz_kernel(float* __restrict__ inout, const int* __restrict__ seg,
            const unsigned* __restrict__ smax, float* __restrict__ denom,
            int n) {
    const int i    = blockIdx.x * blockDim.x + threadIdx.x;
    const int lane = threadIdx.x & 31;
    const bool valid = (i < n);
    int   s = valid ? seg[i] : -1;
    float z = 0.0f;
    if (valid) {
        float m = o2f(smax[s]);
        z = expf(inout[i] - m);
        inout[i] = z;
    }
    float v = z;
#pragma unroll
    for (int d = 1; d < 32; d <<= 1) {
        float ov = __shfl_down(v, (unsigned)d, 32);
        int   os = __shfl_down(s, (unsigned)d, 32);
        if ((lane + d < 32) && (os == s)) v += ov;
    }
    int  ps   = __shfl_up(s, 1u, 32);
    bool head = (lane == 0) || (ps != s);
    if (head && valid) atomicAdd(&denom[s], v);
}

// ---- pass 4: normalize ------------------------------------------------------
__global__ __launch_bounds__(256) void
norm_kernel(float* __restrict__ inout, const int* __restrict__ seg,
            const float* __restrict__ denom, int n) {
    int i = blockIdx.x * blockDim.x + threadIdx.x;
    if (i >= n) return;
    inout[i] = inout[i] / denom[seg[i]];
}

extern "C" void kernel_launch(void* const* d_in, const int* in_sizes, int n_in,
                              void* d_out, int out_size, void* d_ws, size_t ws_size,
                              hipStream_t stream) {
    const float* X   = (const float*)d_in[0];
    const int*   seg = (const int*)d_in[1];
    const float* w   = (const float*)d_in[2];
    const float* b   = (const float*)d_in[3];
    float* out = (float*)d_out;

    const int n    = in_sizes[1];   // N rows (seg_ids length)
    const int nseg = NSEG;

    unsigned* smax  = (unsigned*)d_ws;
    float*    denom = (float*)((char*)d_ws + (size_t)nseg * sizeof(unsigned));

    init_seg_kernel<<<(nseg + 255) / 256, 256, 0, stream>>>(smax, denom, nseg);

    const int ntiles         = (n + 15) / 16;
    const int tiles_per_blk  = WAVES_PER_BLOCK * TILES_PER_WAVE;
    const int blocksA        = (ntiles + tiles_per_blk - 1) / tiles_per_blk;
    logits_wmma_kernel<<<blocksA, 256, 0, stream>>>(X, w, b, out, n);

    const int blocksE = (n + 255) / 256;
    seg_max_kernel<<<blocksE, 256, 0, stream>>>(out, seg, smax, n);
    expz_kernel<<<blocksE, 256, 0, stream>>>(out, seg, smax, denom, n);
    norm_kernel<<<blocksE, 256, 0, stream>>>(out, seg, denom, n);
}